// HiddenSoftNGram_40123584479390
// MI455X (gfx1250) — compile-verified
//
#include <hip/hip_runtime.h>
#include <hip/hip_bf16.h>

// Problem constants (from reference): N=3, V=50257, bsz=8, L=4096, d=1024
#define LSEQ  4096
#define DDIM  1024
#define VOCAB 50257
#define NBAT  8
#define KLEN  4094   // L-1 - N + 2
#define KSPLIT 2
#define KCHUNK (DDIM / KSPLIT)           // 512
#define SSTRIDE ((size_t)NBAT * LSEQ)    // one partial buffer, in floats

typedef __attribute__((ext_vector_type(2))) float v2f;
typedef __attribute__((ext_vector_type(8))) float v8f;

// ---------------------------------------------------------------------------
// Kernel 1: for every row t of hidden[b] compute (over this wave's K-slice)
//   dA[b,t] += hidden[b,t] . hidden[b,L-2]
//   dB[b,t] += hidden[b,t] . hidden[b,L-1]
//   sq[b,t] += ||hidden[b,t]||^2
// via V_WMMA_F32_16X16X4_F32. One wave = one 16-row tile x one K half.
// Partials land in per-slice buffers (deterministic combine in kernel 2).
//
// A fragment (16x4 f32, ISA 7.12.2): lane m (m<16) holds row m, K={k0,k0+1};
// lane m+16 holds row m, K={k0+2,k0+3}.
// B fragment (4x16 f32, mirrored): lanes 0..15 hold K={k0,k0+1} of column
// n=lane; lanes 16..31 hold K={k0+2,k0+3} of column n=lane-16.
// Column 0 = q1 = hidden[b,L-2], column 1 = q2 = hidden[b,L-1], rest zero
// (branch-free select keeps EXEC all-ones through the loop).
// D (16x16 f32): lanes 0..15: N=lane, VGPR v -> M=v; lanes 16..31: M=v+8.
// ---------------------------------------------------------------------------
__global__ __launch_bounds__(256) void rowdot_wmma(const float* __restrict__ hidden,
                                                   float* __restrict__ dA,
                                                   float* __restrict__ dB,
                                                   float* __restrict__ sq) {
  const int lane   = threadIdx.x & 31;
  const int wave   = threadIdx.x >> 5;
  const int ksplit = blockIdx.y;               // 0 or 1: K half
  const int b      = blockIdx.z;
  const int tile   = blockIdx.x * 8 + wave;    // 0..255 (256 tiles of 16 rows)
  const int row0   = tile << 4;
  const int kbeg   = ksplit * KCHUNK;
  const int kend   = kbeg + KCHUNK;

  const float* __restrict__ H    = hidden + (size_t)b * LSEQ * DDIM;
  const float* __restrict__ arow = H + (size_t)(row0 + (lane & 15)) * DDIM;
  const int   khalf = (lane >> 4) << 1;        // 0 for lanes 0-15, 2 for 16-31
  const int   ncol  = lane & 15;
  const float* __restrict__ qrow = H + (size_t)(LSEQ - 2 + (ncol & 1)) * DDIM;
  const bool  bact  = (ncol < 2);              // only columns 0,1 carry data

  v8f   c = {0.f, 0.f, 0.f, 0.f, 0.f, 0.f, 0.f, 0.f};
  float sqx = 0.f, sqy = 0.f;                  // two chains (break FMA dep)

  #pragma unroll 8
  for (int k0 = kbeg; k0 < kend; k0 += 4) {
    // Streamed exactly once per element across the grid -> non-temporal.
    v2f a = __builtin_nontemporal_load((const v2f*)(arow + k0 + khalf));
    // Unconditional load (lanes n>=2 re-read the hot q rows), then select-0.
    v2f bq = *(const v2f*)(qrow + k0 + khalf);
    v2f bb;
    bb.x = bact ? bq.x : 0.f;
    bb.y = bact ? bq.y : 0.f;
    // 8-arg form: (neg_a, A, neg_b, B, c_mod, C, reuse_a, reuse_b)
    c = __builtin_amdgcn_wmma_f32_16x16x4_f32(false, a, false, bb,
                                              (short)0, c, false, false);
    sqx = fmaf(a.x, a.x, sqx);
    sqy = fmaf(a.y, a.y, sqy);
  }

  // Row-norm halves: lane m covered K%4 in {0,1}, lane m+16 covered {2,3}.
  float sqacc = sqx + sqy;
  sqacc += __shfl_xor(sqacc, 16, 32);

  const size_t slice = (size_t)ksplit * SSTRIDE;
  if (lane < 16) sq[slice + (size_t)b * LSEQ + row0 + lane] = sqacc;

  float* dAo = dA + slice + (size_t)b * LSEQ + row0;
  float* dBo = dB + slice + (size_t)b * LSEQ + row0;
  if (lane == 0)  { for (int m = 0; m < 8; ++m) dAo[m]     = c[m]; }
  if (lane == 16) { for (int m = 0; m < 8; ++m) dAo[8 + m] = c[m]; }
  if (lane == 1)  { for (int m = 0; m < 8; ++m) dBo[m]     = c[m]; }
  if (lane == 17) { for (int m = 0; m < 8; ++m) dBo[8 + m] = c[m]; }
}

// ---------------------------------------------------------------------------
// Kernel 2: combine K-slice partials, score -> clamp -> scatter-max.
// clip(max_k(s_k, 0), 0, 1) == max_k(clip(s_k,0,1), 0): clamp per score and
// atomicMax on the uint bit pattern (monotone for non-negative floats).
// ---------------------------------------------------------------------------
__global__ void score_scatter(const float* __restrict__ dA,
                              const float* __restrict__ dB,
                              const float* __restrict__ sq,
                              const int*  __restrict__ prefix,
                              float* __restrict__ out) {
  const int k = blockIdx.x * blockDim.x + threadIdx.x;
  const int b = blockIdx.y;
  if (k >= KLEN) return;
  const size_t base = (size_t)b * LSEQ;
  const float sqk0 = sq[base + k]            + sq[SSTRIDE + base + k];
  const float sqk1 = sq[base + k + 1]        + sq[SSTRIDE + base + k + 1];
  const float sqq0 = sq[base + LSEQ - 2]     + sq[SSTRIDE + base + LSEQ - 2];
  const float sqq1 = sq[base + LSEQ - 1]     + sq[SSTRIDE + base + LSEQ - 1];
  const float num  = (dA[base + k]     + dA[SSTRIDE + base + k]) +
                     (dB[base + k + 1] + dB[SSTRIDE + base + k + 1]);
  const float qn = fmaxf(sqrtf(sqq0 + sqq1), 1e-12f);
  const float kn = fmaxf(sqrtf(sqk0 + sqk1), 1e-12f);
  float s = num / (qn * kn);
  s = fminf(fmaxf(s, 0.f), 1.f);
  const int v = prefix[base + k + 2];               // idx = prefix[:, N-1:]
  atomicMax((unsigned int*)(out + (size_t)b * VOCAB + v), __float_as_uint(s));
}

__global__ void zero_out(float* __restrict__ out, int n) {
  const int i = blockIdx.x * blockDim.x + threadIdx.x;
  if (i < n) out[i] = 0.f;
}

__global__ void apply_weights(float* __restrict__ out,
                              const float* __restrict__ bd,
                              const float* __restrict__ sw) {
  const int i = blockIdx.x * blockDim.x + threadIdx.x;
  if (i < NBAT * VOCAB) {
    const int v = i % VOCAB;
    out[i] = out[i] * bd[v] * sw[v];
  }
}

// ---------------------------------------------------------------------------
extern "C" void kernel_launch(void* const* d_in, const int* in_sizes, int n_in,
                              void* d_out, int out_size, void* d_ws, size_t ws_size,
                              hipStream_t stream) {
  const int*   prefix = (const int*)d_in[0];    // (8, 4096) token ids
  const float* hidden = (const float*)d_in[1];  // (8, 4096, 1024) f32
  const float* bd     = (const float*)d_in[2];  // (V,) f32
  const float* sw     = (const float*)d_in[3];  // (V,) f32
  float* out = (float*)d_out;                   // (8, V) f32

  // Workspace: dA[2] | dB[2] | sq[2], each slice 8*4096 floats (768 KB total)
  float* dA = (float*)d_ws;                     // [2][8][4096]
  float* dB = dA + KSPLIT * SSTRIDE;            // [2][8][4096]
  float* sq = dB + KSPLIT * SSTRIDE;            // [2][8][4096]

  const int nOut = NBAT * VOCAB;
  zero_out<<<(nOut + 255) / 256, 256, 0, stream>>>(out, nOut);

  dim3 g1(32, KSPLIT, NBAT);            // 32 blks x 8 waves = 256 tiles/batch
  rowdot_wmma<<<g1, 256, 0, stream>>>(hidden, dA, dB, sq);

  dim3 g2((KLEN + 255) / 256, NBAT);
  score_scatter<<<g2, 256, 0, stream>>>(dA, dB, sq, prefix, out);

  apply_weights<<<(nOut + 255) / 256, 256, 0, stream>>>(out, bd, sw);
}